// Attention_76879914598815
// MI455X (gfx1250) — compile-verified
//
#include <hip/hip_runtime.h>
#include <math.h>

// ---------------------------------------------------------------------------
// GQA attention with RoPE + sliding window, fp32 throughout, CDNA5 WMMA
// (V_WMMA_F32_16X16X4_F32) for every matmul.
// GEMMs stage tiles through LDS with the gfx1250 async-to-LDS DMA path
// (GLOBAL_LOAD_ASYNC_TO_LDS_B128 + s_wait_asynccnt), double-buffered,
// cutting L2 traffic ~2.7x vs direct-to-register operand fetch.
// Workspace layout (floats): Q[16M] K[4M] V[4M] Z[16M]  (160 MB total).
// ---------------------------------------------------------------------------

typedef __attribute__((ext_vector_type(2))) float v2f;
typedef __attribute__((ext_vector_type(8))) float v8f;
typedef int v4i __attribute__((vector_size(16)));            // matches builtin
typedef __attribute__((address_space(1))) v4i* gp128_t;      // global v4i*
typedef __attribute__((address_space(3))) v4i* lp128_t;      // LDS v4i*

#define NB    4      // batch
#define LQ    2048   // sequence length
#define EE    2048   // embed dim
#define HQ    16     // query heads
#define HKV   4      // kv heads
#define DIMH  128    // head dim
#define WIN   1024   // sliding window
#define SCALE 0.08838834764831845f  // 1/sqrt(128)

#define KC    32     // k-chunk staged per double-buffer step
#define KCP   36     // padded LDS row stride (mult of 4 -> aligned b128 writes;
                     // 36*lane is injective mod 64 -> conflict-free ds_load_b64)

__device__ __forceinline__ v8f wmma_f32_k4(v2f a, v2f b, v8f c) {
  // D = A(16x4) * B(4x16) + C(16x16), full fp32
  return __builtin_amdgcn_wmma_f32_16x16x4_f32(false, a, false, b, (short)0, c,
                                               false, false);
}

// Async global->LDS DMA, 16B per lane (512B per wave-instruction), ASYNCcnt.
__device__ __forceinline__ void async_b128(const float* g, float* l) {
#if __has_builtin(__builtin_amdgcn_global_load_async_to_lds_b128)
  __builtin_amdgcn_global_load_async_to_lds_b128(
      (gp128_t)(v4i*)g, (lp128_t)(v4i*)l, 0, 0);
#else
  unsigned las = (unsigned)(unsigned long long)
      (__attribute__((address_space(3))) char*)l;
  asm volatile("global_load_async_to_lds_b128 %0, %1, off"
               :: "v"(las), "v"(g) : "memory");
#endif
}

__device__ __forceinline__ void wait_async0() {
#if __has_builtin(__builtin_amdgcn_s_wait_asynccnt)
  __builtin_amdgcn_s_wait_asynccnt(0);
#else
  asm volatile("s_wait_asynccnt 0x0" ::: "memory");
#endif
}

// ---------------------------------------------------------------------------
// C = A * W^T WMMA GEMM with async-LDS double-buffered staging.
// A: MxK row-major, W: Ncols x K row-major. Block = 256 threads (8 waves)
// computing a 64x128 tile; each wave owns a 32x32 sub-tile (2x2 WMMA accums).
// mode 0: row-major out; mode 1: scatter into (N, HH, L, 128) head-major.
// ---------------------------------------------------------------------------
__global__ __launch_bounds__(256)
void gemm_xwT(const float* __restrict__ A, const float* __restrict__ W,
              float* __restrict__ C, int M, int Ncols, int K, int mode, int HH)
{
  __shared__ float Abuf[2][64 * KCP];    //  9216 B each
  __shared__ float Bbuf[2][128 * KCP];   // 18432 B each
  const int lane = threadIdx.x & 31;
  const int wid  = threadIdx.x >> 5;
  const int lm = lane & 15, lh = lane >> 4;
  const int row0 = blockIdx.x * 64;
  const int col0 = blockIdx.y * 128;
  const int wrow = (wid & 1) * 32;       // wave sub-tile origin inside block
  const int wcol = (wid >> 1) * 32;
  const int nk = K / KC;

  // ---- stage one k-chunk (A: 512 b128 chunks, B: 1024) -------------------
  auto stage = [&](int buf, int k0) {
#pragma unroll
    for (int i = 0; i < 2; ++i) {
      const int id  = (wid * 2 + i) * 32 + lane;
      const int row = id >> 3, kp = (id & 7) << 2;
      async_b128(A + (size_t)(row0 + row) * K + k0 + kp,
                 &Abuf[buf][row * KCP + kp]);
    }
#pragma unroll
    for (int i = 0; i < 4; ++i) {
      const int id  = (wid * 4 + i) * 32 + lane;
      const int row = id >> 3, kp = (id & 7) << 2;
      async_b128(W + (size_t)(col0 + row) * K + k0 + kp,
                 &Bbuf[buf][row * KCP + kp]);
    }
  };

  const v8f z8 = {0.f, 0.f, 0.f, 0.f, 0.f, 0.f, 0.f, 0.f};
  v8f acc[2][2];
  acc[0][0] = z8; acc[0][1] = z8; acc[1][0] = z8; acc[1][1] = z8;

  stage(0, 0);

  for (int kc = 0; kc < nk; ++kc) {
    const int cur = kc & 1;
    wait_async0();        // my async loads for 'cur' have landed in LDS
    __syncthreads();      // everyone's loads landed; prev compute done

    if (kc + 1 < nk) stage(cur ^ 1, (kc + 1) * KC);

    const float* ap0 = &Abuf[cur][(wrow + lm) * KCP + 2 * lh];
    const float* ap1 = ap0 + 16 * KCP;
    const float* bp0 = &Bbuf[cur][(wcol + lm) * KCP + 2 * lh];
    const float* bp1 = bp0 + 16 * KCP;

#pragma unroll
    for (int k = 0; k < KC; k += 4) {
      v2f a0 = *(const v2f*)(ap0 + k);
      v2f a1 = *(const v2f*)(ap1 + k);
      v2f b0 = *(const v2f*)(bp0 + k);
      v2f b1 = *(const v2f*)(bp1 + k);
      acc[0][0] = wmma_f32_k4(a0, b0, acc[0][0]);
      acc[0][1] = wmma_f32_k4(a0, b1, acc[0][1]);
      acc[1][0] = wmma_f32_k4(a1, b0, acc[1][0]);
      acc[1][1] = wmma_f32_k4(a1, b1, acc[1][1]);
    }
    __syncthreads();      // done reading 'cur' before it is restaged
  }

#pragma unroll
  for (int i = 0; i < 2; ++i) {
#pragma unroll
    for (int j = 0; j < 2; ++j) {
#pragma unroll
      for (int r = 0; r < 8; ++r) {
        const int m = row0 + wrow + i * 16 + r + 8 * lh;
        const int c = col0 + wcol + j * 16 + lm;
        const float v = acc[i][j][r];
        if (mode == 0) {
          C[(size_t)m * Ncols + c] = v;
        } else {
          const int bb = m >> 11;          // m / LQ
          const int ll = m & (LQ - 1);
          const int hh = c >> 7;           // c / DIMH
          const int dd = c & (DIMH - 1);
          C[(((size_t)(bb * HH + hh) << 11) + ll) * DIMH + dd] = v;
        }
      }
    }
  }
}

// ---------------------------------------------------------------------------
// In-place RoPE on a (N*HH, L, 128) tensor; one thread per (row, pair).
// ---------------------------------------------------------------------------
__global__ void rope_kernel(float* __restrict__ T, long total)
{
  const long idx = (long)blockIdx.x * blockDim.x + threadIdx.x;
  if (idx >= total) return;
  const int  t  = (int)(idx & 63);
  const int  l  = (int)((idx >> 6) & (LQ - 1));
  const long bh = idx >> 17;                 // / (64 * LQ)
  // freq = 10000^(-2t/128) = exp(-t * ln(10000)/64)
  const float freq = __expf(-0.14391156632f * (float)t);
  const float ang  = (float)l * freq;
  float sv, cv;
  __sincosf(ang, &sv, &cv);
  float* p = T + ((bh << 11) + l) * (long)DIMH + 2 * t;
  const float xr = p[0], xi = p[1];
  p[0] = xr * cv - xi * sv;
  p[1] = xr * sv + xi * cv;
}

// ---------------------------------------------------------------------------
// Flash attention, sliding-window causal. One wave per 16-query tile.
// Q A-operands preloaded (64 VGPRs). Per 16-key tile: 32 WMMA (S = Q K^T),
// register online softmax (shfl_xor within 16-lane groups matches the WMMA
// C-layout), P re-layout via per-wave LDS bounce, 32 WMMA (O += P V).
// ---------------------------------------------------------------------------
__global__ __launch_bounds__(128)
void attn_kernel(const float* __restrict__ Q, const float* __restrict__ Kx,
                 const float* __restrict__ Vx, float* __restrict__ Z)
{
  __shared__ float Pst[4][16 * 18];          // per-wave P staging, even stride
  const int lane = threadIdx.x & 31;
  const int wid  = threadIdx.x >> 5;
  const int lm = lane & 15, lh = lane >> 4;

  const int gid = blockIdx.x * 4 + wid;      // q-tile id, uniform per wave
  const int qt  = gid & 127;                 // L/16 = 128
  const int h   = (gid >> 7) & 15;
  const int b   = gid >> 11;
  const int q0  = qt << 4;
  const int hk  = h >> 2;                    // GQA group of 4

  // Preload Q tile as WMMA A-operands: 32 k-steps of float2 per lane.
  const float* qrow =
      Q + (((size_t)(b * HQ + h) << 11) + q0 + lm) * DIMH + 2 * lh;
  v2f qa[32];
#pragma unroll
  for (int t = 0; t < 32; ++t) qa[t] = *(const v2f*)(qrow + 4 * t);

  const float* Kb = Kx + (((size_t)(b * HKV + hk)) << 11) * DIMH;
  const float* Vb = Vx + (((size_t)(b * HKV + hk)) << 11) * DIMH;

  const v8f z8 = {0.f, 0.f, 0.f, 0.f, 0.f, 0.f, 0.f, 0.f};
  v8f O[8];
#pragma unroll
  for (int i = 0; i < 8; ++i) O[i] = z8;
  float mrow[8], lrow[8];
#pragma unroll
  for (int r = 0; r < 8; ++r) { mrow[r] = -__builtin_inff(); lrow[r] = 0.f; }

  float* pw = &Pst[wid][0];
  const int jt_lo = (q0 >= WIN) ? ((q0 - (WIN - 1)) >> 4) : 0;

  for (int jt = jt_lo; jt <= qt; ++jt) {
    const int key0 = jt << 4;

    // S = Q * K^T  (K rows are contiguous -> float2 B-operand loads)
    v8f S = z8;
    const float* krow = Kb + (size_t)(key0 + lm) * DIMH + 2 * lh;
#pragma unroll
    for (int t = 0; t < 32; ++t) {
      v2f kb = *(const v2f*)(krow + 4 * t);
      S = wmma_f32_k4(qa[t], kb, S);
    }

    // mask + online softmax; C-layout: VGPR r holds rows (r | r+8), col = lm
    const int n_abs = key0 + lm;
    float alphas[8];
#pragma unroll
    for (int r = 0; r < 8; ++r) {
      const int m_abs = q0 + r + 8 * lh;
      const bool ok = (n_abs <= m_abs) && (n_abs + WIN > m_abs);
      float s = ok ? S[r] * SCALE : -__builtin_inff();
      float tm = s;
      tm = fmaxf(tm, __shfl_xor(tm, 1));
      tm = fmaxf(tm, __shfl_xor(tm, 2));
      tm = fmaxf(tm, __shfl_xor(tm, 4));
      tm = fmaxf(tm, __shfl_xor(tm, 8));
      const float mnew  = fmaxf(mrow[r], tm);
      const float alpha = (mrow[r] == -__builtin_inff())
                              ? 0.f : __expf(mrow[r] - mnew);
      float p = (s == -__builtin_inff()) ? 0.f : __expf(s - mnew);
      float ps = p;
      ps += __shfl_xor(ps, 1);
      ps += __shfl_xor(ps, 2);
      ps += __shfl_xor(ps, 4);
      ps += __shfl_xor(ps, 8);
      lrow[r]  = lrow[r] * alpha + ps;
      mrow[r]  = mnew;
      alphas[r] = alpha;
      pw[(r + 8 * lh) * 18 + lm] = p;        // store P row-major to LDS
    }

    // Rescale accumulators (stats layout == accumulator layout)
#pragma unroll
    for (int dt = 0; dt < 8; ++dt)
#pragma unroll
      for (int r = 0; r < 8; ++r)
        O[dt][r] *= alphas[r];

    // Reload P as WMMA A-operands (aligned ds_load_b64, same-wave in-order)
    v2f pa[4];
#pragma unroll
    for (int j = 0; j < 4; ++j)
      pa[j] = *(const v2f*)(pw + lm * 18 + 4 * j + 2 * lh);

    // O += P * V
#pragma unroll
    for (int dt = 0; dt < 8; ++dt) {
      const float* vcol =
          Vb + (size_t)(key0 + 2 * lh) * DIMH + dt * 16 + lm;
#pragma unroll
      for (int ks = 0; ks < 4; ++ks) {
        v2f vb;
        vb.x = vcol[(4 * ks) * DIMH];
        vb.y = vcol[(4 * ks + 1) * DIMH];
        O[dt] = wmma_f32_k4(pa[ks], vb, O[dt]);
      }
    }
  }

  // Normalize and scatter into Z (N, L, E) for the output projection.
  float rdiv[8];
#pragma unroll
  for (int r = 0; r < 8; ++r) rdiv[r] = 1.0f / lrow[r];

#pragma unroll
  for (int dt = 0; dt < 8; ++dt)
#pragma unroll
    for (int r = 0; r < 8; ++r) {
      const int m = q0 + r + 8 * lh;
      const int c = h * DIMH + dt * 16 + lm;
      Z[(((size_t)b << 11) + m) * EE + c] = O[dt][r] * rdiv[r];
    }
}

// ---------------------------------------------------------------------------
extern "C" void kernel_launch(void* const* d_in, const int* in_sizes, int n_in,
                              void* d_out, int out_size, void* d_ws,
                              size_t ws_size, hipStream_t stream)
{
  (void)in_sizes; (void)n_in; (void)out_size; (void)ws_size;
  const float* x  = (const float*)d_in[0];
  const float* Wq = (const float*)d_in[1];
  const float* Wk = (const float*)d_in[2];
  const float* Wv = (const float*)d_in[3];
  const float* Wo = (const float*)d_in[4];
  float* out = (float*)d_out;

  float* Qw = (float*)d_ws;                                   // 16M floats
  float* Kw = Qw + (size_t)NB * HQ  * LQ * DIMH;              //  4M floats
  float* Vw = Kw + (size_t)NB * HKV * LQ * DIMH;              //  4M floats
  float* Zw = Vw + (size_t)NB * HKV * LQ * DIMH;              // 16M floats

  const int M = NB * LQ;                                      // 8192
  dim3 blk(256);

  // QKV projections (head-major epilogue)
  gemm_xwT<<<dim3(M / 64, EE / 128),        blk, 0, stream>>>(
      x, Wq, Qw, M, EE,      EE, 1, HQ);
  gemm_xwT<<<dim3(M / 64, (EE / 4) / 128),  blk, 0, stream>>>(
      x, Wk, Kw, M, EE / 4,  EE, 1, HKV);
  gemm_xwT<<<dim3(M / 64, (EE / 4) / 128),  blk, 0, stream>>>(
      x, Wv, Vw, M, EE / 4,  EE, 1, HKV);

  // RoPE on Q and K
  const long pq = (long)NB * HQ * LQ * 64;
  rope_kernel<<<(unsigned)((pq + 255) / 256), 256, 0, stream>>>(Qw, pq);
  const long pk = (long)NB * HKV * LQ * 64;
  rope_kernel<<<(unsigned)((pk + 255) / 256), 256, 0, stream>>>(Kw, pk);

  // Flash attention: 8192 q-tiles, 4 waves per block
  attn_kernel<<<(NB * HQ * (LQ / 16)) / 4, 128, 0, stream>>>(Qw, Kw, Vw, Zw);

  // Output projection (row-major epilogue straight into d_out)
  gemm_xwT<<<dim3(M / 64, EE / 128), blk, 0, stream>>>(
      Zw, Wo, out, M, EE, EE, 0, 0);
}